// MMSE_34978213658818
// MI455X (gfx1250) — compile-verified
//
#include <hip/hip_runtime.h>
#include <cstdint>

#define N_USERS   50000
#define DEG       32
#define N_EDGES   (N_USERS * DEG)
#define N_SERVERS 512

// CDNA5 async global->LDS path (CDNA5 ISA), guarded so
// compilation never fails if the toolchain spells the builtin differently.
#if defined(__AMDGCN__) && __has_builtin(__builtin_amdgcn_global_load_async_to_lds_b32)
#define USE_ASYNC_LDS 1
#else
#define USE_ASYNC_LDS 0
#endif

__device__ __forceinline__ float wave_reduce_max(float v) {
#pragma unroll
  for (int m = 16; m > 0; m >>= 1) v = fmaxf(v, __shfl_xor(v, m, 32));
  return v;
}

__device__ __forceinline__ float wave_reduce_sum(float v) {
#pragma unroll
  for (int m = 16; m > 0; m >>= 1) v += __shfl_xor(v, m, 32);
  return v;
}

// ---------------------------------------------------------------------------
// Kernel 0: zero the scratch (512 server power bins + 1 loss accumulator).
// ---------------------------------------------------------------------------
__global__ void k_zero(float* __restrict__ ws) {
  int i = blockIdx.x * blockDim.x + threadIdx.x;
  if (i < N_SERVERS + 1) ws[i] = 0.0f;
}

// ---------------------------------------------------------------------------
// Kernel 1: per-user power softmax (one wave32 per user, lane == edge-in-user),
// pw = power_sche * path_loss, accumulate per-server power with f32 atomics.
// ---------------------------------------------------------------------------
__global__ __launch_bounds__(256) void k_server_power(
    const float* __restrict__ power_alloc,
    const float* __restrict__ path_losses,
    const int* __restrict__ edge_index,   // [2*E]; row 1 = server index
    float* __restrict__ server_pw) {
  const int u    = blockIdx.x * 8 + (threadIdx.x >> 5);  // 8 users per block
  const int lane = threadIdx.x & 31;
  const int e    = (u << 5) + lane;                      // coalesced edges

  float pa = power_alloc[e];
  float m  = wave_reduce_max(pa);
  float ex = __expf(pa - m);
  float s  = wave_reduce_sum(ex) + 1e-16f;
  float pw = (ex / s) * path_losses[e];

  int sv = edge_index[N_EDGES + e];
  atomicAdd(&server_pw[sv], pw);
}

// ---------------------------------------------------------------------------
// Kernel 2: full edge model. Stage the 512 per-server sums into LDS with the
// CDNA5 async tensor-data path, recompute the three wave32 softmaxes, compute
// rate/time-loss, wave-sum -> block-sum -> one global atomic per block.
// ---------------------------------------------------------------------------
__global__ __launch_bounds__(256) void k_loss(
    const float* __restrict__ compute_resource,  // [512]
    const float* __restrict__ path_losses,       // [E]
    const float* __restrict__ task_size,         // [N_USERS]
    const int* __restrict__ edge_index,          // [2*E]
    const float* __restrict__ task_alloc,        // [E]
    const float* __restrict__ power_alloc,       // [E]
    const float* __restrict__ comp_alloc,        // [E]
    const float* __restrict__ server_pw,         // [512] (ws)
    float* __restrict__ acc) {                   // [1]   (ws)
  __shared__ float s_pws[N_SERVERS];
  __shared__ float s_part[8];

  const int t = threadIdx.x;  // 256 threads stage 512 floats (2 each)
#if USE_ASYNC_LDS
  {
    typedef __attribute__((address_space(1))) int gint;  // __device__ int*
    typedef __attribute__((address_space(3))) int lint;  // __shared__ int*
    // Explicit pointers for both halves: no reliance on how the builtin's
    // immediate offset is applied (ISA adds IOFFSET to both sides, but we
    // cannot verify the intrinsic's lowering at runtime in this loop).
    gint* g0 = (gint*)(server_pw + t);
    gint* g1 = (gint*)(server_pw + t + 256);
    lint* l0 = (lint*)(&s_pws[t]);
    lint* l1 = (lint*)(&s_pws[t + 256]);
    __builtin_amdgcn_global_load_async_to_lds_b32(g0, l0, 0, 0);
    __builtin_amdgcn_global_load_async_to_lds_b32(g1, l1, 0, 0);
#if __has_builtin(__builtin_amdgcn_s_wait_asynccnt)
    __builtin_amdgcn_s_wait_asynccnt(0);
#else
    asm volatile("s_wait_asynccnt 0" ::: "memory");
#endif
  }
#else
  s_pws[t]       = server_pw[t];
  s_pws[t + 256] = server_pw[t + 256];
#endif
  __syncthreads();

  const int u    = blockIdx.x * 8 + (t >> 5);
  const int lane = t & 31;
  const int e    = (u << 5) + lane;
  const int wv   = t >> 5;

  float ta = task_alloc[e];
  float ca = comp_alloc[e];
  float pa = power_alloc[e];

  // three segment softmaxes, each a wave32 butterfly (segment == wave)
  float tm = wave_reduce_max(ta);
  float te = __expf(ta - tm);
  float ts = wave_reduce_sum(te) + 1e-16f;
  float task_sche = te / ts;

  float cm = wave_reduce_max(ca);
  float ce = __expf(ca - cm);
  float cs = wave_reduce_sum(ce) + 1e-16f;
  float comp_sche = ce / cs;

  float pm = wave_reduce_max(pa);
  float pe = __expf(pa - pm);
  float ps = wave_reduce_sum(pe) + 1e-16f;
  float power_sche = pe / ps;

  float pw    = power_sche * path_losses[e];
  int   sv    = edge_index[N_EDGES + e];
  float tasks = task_size[u] * task_sche;
  float comp  = compute_resource[sv] * comp_sche;

  float interference = s_pws[sv] - pw;
  // log2(1+x) via log1p to avoid near-1 cancellation (x ~ 3e-4 here)
  float rate = log1pf(pw / (interference + 1e-9f)) * 1.4426950408889634f;

  float tl = tasks / (rate + 1e-20f) + tasks / (comp + 1e-20f);

  float usum = wave_reduce_sum(tl);       // per-user time loss
  if (lane == 0) s_part[wv] = usum;
  __syncthreads();
  if (t == 0) {
    float b = 0.0f;
#pragma unroll
    for (int i = 0; i < 8; ++i) b += s_part[i];
    atomicAdd(acc, b);                    // one atomic per 8 users
  }
}

// ---------------------------------------------------------------------------
// Kernel 3: finalize mean.
// ---------------------------------------------------------------------------
__global__ void k_finalize(const float* __restrict__ acc, float* __restrict__ out) {
  if (threadIdx.x == 0 && blockIdx.x == 0)
    out[0] = acc[0] * (1.0f / (float)N_USERS);
}

extern "C" void kernel_launch(void* const* d_in, const int* in_sizes, int n_in,
                              void* d_out, int out_size, void* d_ws, size_t ws_size,
                              hipStream_t stream) {
  const float* compute_resource = (const float*)d_in[0];
  const float* path_losses      = (const float*)d_in[1];
  const float* task_size        = (const float*)d_in[2];
  const int*   edge_index       = (const int*)d_in[3];
  const float* task_alloc       = (const float*)d_in[4];
  const float* power_alloc      = (const float*)d_in[5];
  const float* comp_alloc       = (const float*)d_in[6];

  float* ws        = (float*)d_ws;
  float* server_pw = ws;               // [512]
  float* acc       = ws + N_SERVERS;   // [1]
  float* out       = (float*)d_out;

  k_zero<<<3, 256, 0, stream>>>(ws);
  k_server_power<<<N_USERS / 8, 256, 0, stream>>>(power_alloc, path_losses,
                                                  edge_index, server_pw);
  k_loss<<<N_USERS / 8, 256, 0, stream>>>(compute_resource, path_losses, task_size,
                                          edge_index, task_alloc, power_alloc,
                                          comp_alloc, server_pw, acc);
  k_finalize<<<1, 32, 0, stream>>>(acc, out);
}